// GCNNetwork_37641093382622
// MI455X (gfx1250) — compile-verified
//
#include <hip/hip_runtime.h>
#include <hip/hip_bf16.h>

typedef __attribute__((ext_vector_type(2))) float v2f;
typedef __attribute__((ext_vector_type(8))) float v8f;

#define HID 64
#define IN_C 128
#define OUT_C 40

// ---------------------------------------------------------------------------
// Degree / normalization
// ---------------------------------------------------------------------------
__global__ void init_deg_kernel(float* deg, int n) {
    int i = blockIdx.x * blockDim.x + threadIdx.x;
    if (i < n) deg[i] = 1.0f;  // self-loop
}

__global__ void count_deg_kernel(const long long* __restrict__ ei, float* deg, int E) {
    int e = blockIdx.x * blockDim.x + threadIdx.x;
    if (e < E) {
        long long d = ei[(size_t)E + e];  // dst row
        unsafeAtomicAdd(&deg[d], 1.0f);
    }
}

__global__ void dinv_kernel(float* deg, int n) {
    int i = blockIdx.x * blockDim.x + threadIdx.x;
    if (i < n) deg[i] = rsqrtf(deg[i]);
}

__global__ void zero_kernel(float* p, size_t n) {
    size_t i = (size_t)blockIdx.x * blockDim.x + threadIdx.x;
    if (i < n) p[i] = 0.0f;
}

// ---------------------------------------------------------------------------
// f32 WMMA GEMM, fully compile-time geometry:
//   C[M x ldc] = A[M x K] @ B[K x LDB] (+bias, +relu), COLMAX valid columns.
// One wave computes a 16-row x (NT*16)-col strip with V_WMMA_F32_16X16X4_F32.
// K/LDB/COLMAX are template params so the K-loop fully unrolls and every
// global load becomes base + immediate offset (no per-step 64-bit address
// arithmetic), and the column mask folds away when COLMAX == NT*16.
//
// A layout (16x4 MxK): lanes 0-15 -> M rows; kHalf=lane>>4 selects K{0,1}/{2,3}.
// B layout (4x16 KxN): lanes 0-15 -> N cols; VGPR j holds K = j + 2*kHalf.
// C/D layout: VGPR j -> row j + 8*(lane>>4), col = lane&15.
// EXEC stays all-ones throughout (WMMA requirement); OOB columns handled
// branchlessly via clamped index + 0/1 mask multiply.
// Grid: (M/16), blockDim = 32 (one full wave).
// ---------------------------------------------------------------------------
template <int NT, int K, int LDB, int COLMAX>
__global__ void gemm_wmma_f32(const float* __restrict__ A, const float* __restrict__ B,
                              const float* __restrict__ bias, float* __restrict__ C,
                              int ldc, int doRelu) {
    const int lane   = threadIdx.x & 31;
    const int lane15 = lane & 15;
    const int kHalf  = lane >> 4;
    const int tileM  = blockIdx.x;
    const int rowA   = tileM * 16 + lane15;

    // Per-tile clamped column + 0/1 mask (folds to identity when COLMAX==NT*16)
    int   colc[NT];
    float cmask[NT];
#pragma unroll
    for (int t = 0; t < NT; ++t) {
        int col  = t * 16 + lane15;
        cmask[t] = (col < COLMAX) ? 1.0f : 0.0f;
        colc[t]  = (col < COLMAX) ? col : (COLMAX - 1);
    }

    v8f acc[NT];
#pragma unroll
    for (int t = 0; t < NT; ++t) acc[t] = (v8f){0.f, 0.f, 0.f, 0.f, 0.f, 0.f, 0.f, 0.f};

    // Single base pointer per stream; all loop accesses at immediate offsets.
    const float* __restrict__ abase = A + (size_t)rowA * K + 2 * kHalf;
    const float* bbase[NT];
#pragma unroll
    for (int t = 0; t < NT; ++t) bbase[t] = B + (size_t)(2 * kHalf) * LDB + colc[t];

#pragma unroll
    for (int k0 = 0; k0 < K; k0 += 4) {
        v2f a = *(const v2f*)(abase + k0);  // global_load_b64, imm offset k0*4
#pragma unroll
        for (int t = 0; t < NT; ++t) {
            v2f b;
            b.x = bbase[t][k0 * LDB]       * cmask[t];  // imm offset
            b.y = bbase[t][(k0 + 1) * LDB] * cmask[t];  // imm offset
            acc[t] = __builtin_amdgcn_wmma_f32_16x16x4_f32(
                /*neg_a=*/false, a, /*neg_b=*/false, b,
                /*c_mod=*/(short)0, acc[t], /*reuse_a=*/false, /*reuse_b=*/false);
        }
    }

#pragma unroll
    for (int t = 0; t < NT; ++t) {
        const int   col = t * 16 + lane15;
        const float bv  = (bias != nullptr) ? bias[colc[t]] * cmask[t] : 0.0f;
        if (col < COLMAX) {
#pragma unroll
            for (int j = 0; j < 8; ++j) {
                int row = tileM * 16 + j + 8 * kHalf;
                float v = acc[t][j] + bv;
                if (doRelu) v = fmaxf(v, 0.0f);
                C[(size_t)row * ldc + col] = v;
            }
        }
    }
}

// ---------------------------------------------------------------------------
// Edge scatter: agg[dst] += dinv[src]*dinv[dst] * hw[src]  (float4 per thread)
// ---------------------------------------------------------------------------
__global__ void scatter_edges_kernel(const long long* __restrict__ ei,
                                     const float* __restrict__ hw,
                                     const float* __restrict__ dinv,
                                     float* agg, int E) {
    size_t t = (size_t)blockIdx.x * blockDim.x + threadIdx.x;
    size_t e = t >> 4;               // 16 chunks of 4 features per edge
    int    c = (int)(t & 15);
    if (e >= (size_t)E) return;
    long long s = ei[e];
    long long d = ei[(size_t)E + e];
    float w = dinv[s] * dinv[d];
    const float4 hv = reinterpret_cast<const float4*>(hw + (size_t)s * HID)[c];  // b128
    float* ad = agg + (size_t)d * HID + c * 4;
    unsafeAtomicAdd(&ad[0], w * hv.x);
    unsafeAtomicAdd(&ad[1], w * hv.y);
    unsafeAtomicAdd(&ad[2], w * hv.z);
    unsafeAtomicAdd(&ad[3], w * hv.w);
}

// ---------------------------------------------------------------------------
// Fused finalize (+self-loop +bias) + LayerNorm + ReLU. One wave per node,
// each lane owns features {lane, lane+32}; wave32 shfl_xor reduction.
// ---------------------------------------------------------------------------
__global__ void finalize_ln_relu_kernel(const float* __restrict__ agg,
                                        const float* __restrict__ hw,
                                        const float* __restrict__ dinv,
                                        const float* __restrict__ bias,
                                        const float* __restrict__ g,
                                        const float* __restrict__ be,
                                        float* __restrict__ out, int n) {
    int node = (int)(((size_t)blockIdx.x * blockDim.x + threadIdx.x) >> 5);
    int lane = threadIdx.x & 31;
    if (node >= n) return;  // uniform per wave
    size_t base = (size_t)node * HID;
    float sl = dinv[node] * dinv[node];
    float v0 = agg[base + lane]      + sl * hw[base + lane]      + bias[lane];
    float v1 = agg[base + lane + 32] + sl * hw[base + lane + 32] + bias[lane + 32];
    float s  = v0 + v1;
    float ss = v0 * v0 + v1 * v1;
#pragma unroll
    for (int m = 16; m >= 1; m >>= 1) {
        s  += __shfl_xor(s,  m, 32);
        ss += __shfl_xor(ss, m, 32);
    }
    float mu  = s * (1.0f / HID);
    float var = ss * (1.0f / HID) - mu * mu;
    float inv = rsqrtf(var + 1e-5f);
    out[base + lane]      = fmaxf((v0 - mu) * inv * g[lane]      + be[lane],      0.0f);
    out[base + lane + 32] = fmaxf((v1 - mu) * inv * g[lane + 32] + be[lane + 32], 0.0f);
}

// Plain finalize (layer 3: no LN / ReLU)
__global__ void finalize_plain_kernel(const float* __restrict__ agg,
                                      const float* __restrict__ hw,
                                      const float* __restrict__ dinv,
                                      const float* __restrict__ bias,
                                      float* __restrict__ out, int n) {
    size_t i = (size_t)blockIdx.x * blockDim.x + threadIdx.x;
    if (i >= (size_t)n * HID) return;
    int node = (int)(i >> 6);
    int c    = (int)(i & 63);
    float sl = dinv[node] * dinv[node];
    out[i] = agg[i] + sl * hw[i] + bias[c];
}

// ---------------------------------------------------------------------------
// Row softmax over OUT_C=40 classes, in place. One thread per row.
// Three passes (max / sum / write) — recompute expf to avoid scratch spills.
// ---------------------------------------------------------------------------
__global__ void softmax_kernel(float* __restrict__ p, int n) {
    int r = blockIdx.x * blockDim.x + threadIdx.x;
    if (r >= n) return;
    float* row = p + (size_t)r * OUT_C;
    float mx = row[0];
#pragma unroll
    for (int j = 1; j < OUT_C; ++j) mx = fmaxf(mx, row[j]);
    float sum = 0.0f;
#pragma unroll
    for (int j = 0; j < OUT_C; ++j) sum += expf(row[j] - mx);
    float rinv = 1.0f / sum;
#pragma unroll
    for (int j = 0; j < OUT_C; ++j) row[j] = expf(row[j] - mx) * rinv;
}

// ---------------------------------------------------------------------------
extern "C" void kernel_launch(void* const* d_in, const int* in_sizes, int n_in,
                              void* d_out, int out_size, void* d_ws, size_t ws_size,
                              hipStream_t stream) {
    const float*     x  = (const float*)d_in[0];
    const long long* ei = (const long long*)d_in[1];
    const float* W1  = (const float*)d_in[2];
    const float* b1  = (const float*)d_in[3];
    const float* W2  = (const float*)d_in[4];
    const float* b2  = (const float*)d_in[5];
    const float* W3  = (const float*)d_in[6];
    const float* b3  = (const float*)d_in[7];
    const float* g1  = (const float*)d_in[8];
    const float* be1 = (const float*)d_in[9];
    const float* g2  = (const float*)d_in[10];
    const float* be2 = (const float*)d_in[11];
    const float* M1  = (const float*)d_in[12];
    const float* mb1 = (const float*)d_in[13];
    const float* M2  = (const float*)d_in[14];
    const float* mb2 = (const float*)d_in[15];

    const int N = in_sizes[0] / IN_C;
    const int E = in_sizes[1] / 2;

    float* ws   = (float*)d_ws;
    float* dinv = ws;                              // N
    float* hw   = ws + N;                          // N*64
    float* agg  = ws + N + (size_t)N * HID;        // N*64
    float* h    = ws + N + (size_t)2 * N * HID;    // N*64
    float* out  = (float*)d_out;                   // N*40

    const size_t NH = (size_t)N * HID;
    const dim3 blk(256);
    const dim3 blk32(32);
    const int  mTiles = N / 16;  // N = 50000 = 3125*16

    // --- degree / dinv ---
    init_deg_kernel<<<(N + 255) / 256, blk, 0, stream>>>(dinv, N);
    count_deg_kernel<<<(E + 255) / 256, blk, 0, stream>>>(ei, dinv, E);
    dinv_kernel<<<(N + 255) / 256, blk, 0, stream>>>(dinv, N);

    const unsigned scatterGrid = (unsigned)(((size_t)E * 16 + 255) / 256);
    const unsigned zeroGrid    = (unsigned)((NH + 255) / 256);

    // --- conv layer 1: x(128) -> h(64), LN+ReLU ---
    gemm_wmma_f32<4, IN_C, HID, HID><<<mTiles, blk32, 0, stream>>>(x, W1, nullptr, hw,
                                                                   HID, 0);
    zero_kernel<<<zeroGrid, blk, 0, stream>>>(agg, NH);
    scatter_edges_kernel<<<scatterGrid, blk, 0, stream>>>(ei, hw, dinv, agg, E);
    finalize_ln_relu_kernel<<<(N * 32 + 255) / 256, blk, 0, stream>>>(agg, hw, dinv,
                                                                      b1, g1, be1, h, N);

    // --- conv layer 2 ---
    gemm_wmma_f32<4, HID, HID, HID><<<mTiles, blk32, 0, stream>>>(h, W2, nullptr, hw,
                                                                  HID, 0);
    zero_kernel<<<zeroGrid, blk, 0, stream>>>(agg, NH);
    scatter_edges_kernel<<<scatterGrid, blk, 0, stream>>>(ei, hw, dinv, agg, E);
    finalize_ln_relu_kernel<<<(N * 32 + 255) / 256, blk, 0, stream>>>(agg, hw, dinv,
                                                                      b2, g2, be2, h, N);

    // --- conv layer 3 (no LN/ReLU) ---
    gemm_wmma_f32<4, HID, HID, HID><<<mTiles, blk32, 0, stream>>>(h, W3, nullptr, hw,
                                                                  HID, 0);
    zero_kernel<<<zeroGrid, blk, 0, stream>>>(agg, NH);
    scatter_edges_kernel<<<scatterGrid, blk, 0, stream>>>(ei, hw, dinv, agg, E);
    finalize_plain_kernel<<<zeroGrid, blk, 0, stream>>>(agg, hw, dinv, b3, h, N);

    // --- MLP layer 1: h @ M1 + mb1, ReLU -> hw ---
    gemm_wmma_f32<4, HID, HID, HID><<<mTiles, blk32, 0, stream>>>(h, M1, mb1, hw,
                                                                  HID, 1);
    // --- MLP layer 2: hw @ M2 + mb2 -> out (40 cols, 3 tiles w/ masked tail) ---
    gemm_wmma_f32<3, HID, OUT_C, OUT_C><<<mTiles, blk32, 0, stream>>>(hw, M2, mb2, out,
                                                                      OUT_C, 0);
    // --- softmax ---
    softmax_kernel<<<(N + 255) / 256, blk, 0, stream>>>(out, N);
}